// Novel_Edge_GCN_90761248899097
// MI455X (gfx1250) — compile-verified
//
#include <hip/hip_runtime.h>
#include <hip/hip_bf16.h>

typedef __attribute__((ext_vector_type(2))) float v2f;
typedef __attribute__((ext_vector_type(8))) float v8f;

#define HID 128
#define CLS 32
#define KEY_NEGINF 0x007FFFFFu   // monotonic key of -inf

// Monotonic float -> uint mapping: unsigned compare orders like float compare.
__device__ __forceinline__ unsigned int f2key(float f) {
    unsigned int b = __float_as_uint(f);
    return (b & 0x80000000u) ? ~b : (b | 0x80000000u);
}
__device__ __forceinline__ float key2f(unsigned int k) {
    unsigned int b = (k & 0x80000000u) ? (k & 0x7FFFFFFFu) : ~k;
    return __uint_as_float(b);
}

// ---------------------------------------------------------------------------
// H[N,128] = X[N,128] @ W[128,128]   (W stored [in][out], row-major)
// 8 waves/block, wave w computes output cols [16w,16w+16). B fragments for the
// whole K=128 cached in 64 VGPRs; per M-tile, all A fragments prefetched from
// LDS into 64 VGPRs (16 ds_load_2addr_b64, ONE dscnt wait), then 32 WMMAs in
// two independent accumulation chains.
// ---------------------------------------------------------------------------
__global__ __launch_bounds__(256) void gemm_h_kernel(
    const float* __restrict__ X, const float* __restrict__ W,
    float* __restrict__ H, int nrows, int mtiles, int mtiles_per_block) {
    __shared__ float xs[16 * HID];
    const int wave = threadIdx.x >> 5;
    const int lane = threadIdx.x & 31;
    const int kh   = lane >> 4;   // which K-half of the 16x16x4 fragment
    const int ln   = lane & 15;   // M index for A, N index for B
    const int n0   = wave * 16;

    // Cache this wave's B slice: B[k][n0+ln] for all k, striped per WMMA layout.
    v2f bfrag[32];
#pragma unroll
    for (int ks = 0; ks < 32; ++ks) {
        const int k = 4 * ks + 2 * kh;
        bfrag[ks].x = W[k * HID + n0 + ln];
        bfrag[ks].y = W[(k + 1) * HID + n0 + ln];
    }

    for (int t = 0; t < mtiles_per_block; ++t) {
        const int mtile = blockIdx.x * mtiles_per_block + t;
        if (mtile >= mtiles) break;
        const int row0 = mtile * 16;
        __syncthreads();   // previous iteration done reading xs
        for (int i = threadIdx.x; i < 16 * HID; i += 256) {
            const int r = row0 + (i >> 7);
            xs[i] = (r < nrows) ? X[row0 * HID + i] : 0.0f;
        }
        __syncthreads();

        // Prefetch ALL A fragments for this tile (single dscnt wait).
        v2f afrag[32];
#pragma unroll
        for (int ks = 0; ks < 32; ++ks) {
            const int k = 4 * ks + 2 * kh;
            afrag[ks].x = xs[ln * HID + k];
            afrag[ks].y = xs[ln * HID + k + 1];
        }

        // Two independent WMMA accumulation chains (even/odd K-steps).
        v8f acc0 = {0.f, 0.f, 0.f, 0.f, 0.f, 0.f, 0.f, 0.f};
        v8f acc1 = {0.f, 0.f, 0.f, 0.f, 0.f, 0.f, 0.f, 0.f};
#pragma unroll
        for (int ks = 0; ks < 32; ks += 2) {
            acc0 = __builtin_amdgcn_wmma_f32_16x16x4_f32(
                false, afrag[ks], false, bfrag[ks], (short)0, acc0, false, false);
            acc1 = __builtin_amdgcn_wmma_f32_16x16x4_f32(
                false, afrag[ks + 1], false, bfrag[ks + 1], (short)0, acc1, false, false);
        }

        // D layout: VGPR r -> row (r + 8*kh), col = n0 + ln
#pragma unroll
        for (int r = 0; r < 8; ++r) {
            const int row = row0 + 8 * kh + r;
            if (row < nrows) H[row * HID + n0 + ln] = acc0[r] + acc1[r];
        }
    }
}

// ---------------------------------------------------------------------------
__global__ void init_agg_kernel(unsigned int* __restrict__ agg, int n4) {
    const int i = blockIdx.x * blockDim.x + threadIdx.x;
    if (i < n4) {
        uint4 v;
        v.x = v.y = v.z = v.w = KEY_NEGINF;
        reinterpret_cast<uint4*>(agg)[i] = v;
    }
}

// ---------------------------------------------------------------------------
// One wave per edge: coalesced 512B gather of h[src], scale by norm, then
// per-lane 4x non-returning u32 atomic max into agg[dst] (L2 atomics).
// Edges >= nE are the self loops (src = dst = e - nE).
// ---------------------------------------------------------------------------
__global__ __launch_bounds__(256) void scatter_max_kernel(
    const float* __restrict__ H, const int* __restrict__ srcs,
    const int* __restrict__ dsts, const float* __restrict__ norm,
    unsigned int* __restrict__ agg, int nE, int nTot) {
    const int wid  = (blockIdx.x * 256 + threadIdx.x) >> 5;
    const int lane = threadIdx.x & 31;
    if (wid >= nTot) return;
    int s, d;
    if (wid < nE) { s = srcs[wid]; d = dsts[wid]; }
    else          { s = wid - nE;  d = s;         }
    const float w = norm[wid];
    const float4 hv = *reinterpret_cast<const float4*>(H + s * HID + lane * 4);
    unsigned int* ap = agg + d * HID + lane * 4;
    atomicMax(ap + 0, f2key(w * hv.x));
    atomicMax(ap + 1, f2key(w * hv.y));
    atomicMax(ap + 2, f2key(w * hv.z));
    atomicMax(ap + 3, f2key(w * hv.w));
}

// ---------------------------------------------------------------------------
__global__ void relu_bias_kernel(const unsigned int* __restrict__ agg,
                                 const float* __restrict__ bias,
                                 float* __restrict__ Xout, int nelem) {
    const int i = blockIdx.x * blockDim.x + threadIdx.x;
    if (i >= nelem) return;
    const unsigned int k = agg[i];
    const float a = (k == KEY_NEGINF) ? 0.0f : key2f(k);   // empty segment -> 0
    const float v = a + bias[i & (HID - 1)];
    Xout[i] = v > 0.0f ? v : 0.0f;
}

// ---------------------------------------------------------------------------
// logits = X @ Wc + bc ; log_softmax along C=32. One wave per row: lane = class.
// ---------------------------------------------------------------------------
__global__ __launch_bounds__(256) void classifier_kernel(
    const float* __restrict__ X, const float* __restrict__ Wc,
    const float* __restrict__ bc, float* __restrict__ out, int nrows) {
    __shared__ float xs[8][HID];
    const int wave = threadIdx.x >> 5;
    const int lane = threadIdx.x & 31;
    const int row  = blockIdx.x * 8 + wave;
    const int rowc = row < nrows ? row : nrows - 1;

    const float4 v = *reinterpret_cast<const float4*>(X + rowc * HID + lane * 4);
    *reinterpret_cast<float4*>(&xs[wave][lane * 4]) = v;
    __syncthreads();

    float acc = bc[lane];
#pragma unroll 8
    for (int k = 0; k < HID; ++k)
        acc = fmaf(xs[wave][k], Wc[k * CLS + lane], acc);

    // log-softmax across the 32 lanes of the wave
    float m = acc;
    for (int off = 16; off > 0; off >>= 1)
        m = fmaxf(m, __shfl_xor(m, off, 32));
    float e = expf(acc - m);
    float s = e;
    for (int off = 16; off > 0; off >>= 1)
        s += __shfl_xor(s, off, 32);
    const float r = acc - m - logf(s);
    if (row < nrows) out[row * CLS + lane] = r;
}

// ---------------------------------------------------------------------------
extern "C" void kernel_launch(void* const* d_in, const int* in_sizes, int n_in,
                              void* d_out, int out_size, void* d_ws, size_t ws_size,
                              hipStream_t stream) {
    (void)n_in; (void)out_size; (void)ws_size;
    const float* x_in = (const float*)d_in[0];
    const int*   ei   = (const int*)d_in[1];
    const float* norm = (const float*)d_in[2];
    const float* Ws   = (const float*)d_in[3];
    const float* bs   = (const float*)d_in[4];
    const float* Wc   = (const float*)d_in[5];
    const float* bc   = (const float*)d_in[6];
    float* out = (float*)d_out;

    const int N = in_sizes[0] / HID;
    const int E = in_sizes[1] / 2;
    const int total_edges = E + N;
    const int nelem = N * HID;

    float* buf_x = (float*)d_ws;
    float* buf_h = buf_x + (size_t)nelem;
    unsigned int* buf_agg = (unsigned int*)(buf_h + (size_t)nelem);

    const int mtiles = (N + 15) / 16;
    const int MPB = 8;
    const int gemm_blocks = (mtiles + MPB - 1) / MPB;
    const int init_blocks = (nelem / 4 + 255) / 256;
    const int scat_blocks = (total_edges + 7) / 8;   // 8 waves per 256-thr block
    const int relu_blocks = (nelem + 255) / 256;
    const int cls_blocks  = (N + 7) / 8;

    const float* cur = x_in;
    for (int l = 0; l < 3; ++l) {
        gemm_h_kernel<<<gemm_blocks, 256, 0, stream>>>(
            cur, Ws + (size_t)l * HID * HID, buf_h, N, mtiles, MPB);
        init_agg_kernel<<<init_blocks, 256, 0, stream>>>(buf_agg, nelem / 4);
        scatter_max_kernel<<<scat_blocks, 256, 0, stream>>>(
            buf_h, ei, ei + E, norm, buf_agg, E, total_edges);
        relu_bias_kernel<<<relu_blocks, 256, 0, stream>>>(
            buf_agg, bs + (size_t)l * HID, buf_x, nelem);
        cur = buf_x;
    }
    classifier_kernel<<<cls_blocks, 256, 0, stream>>>(cur, Wc, bc, out, N);
}